// BFM_11184094838936
// MI455X (gfx1250) — compile-verified
//
#include <hip/hip_runtime.h>
#include <math.h>

typedef __attribute__((ext_vector_type(2))) float v2f;
typedef __attribute__((ext_vector_type(4))) float v4f;
typedef __attribute__((ext_vector_type(8))) float v8f;

#define NV 35709
#define NF 70789
#define BATCH 64
#define M3 (3 * NV)            // 107127 rows of the bases
#define CSTRIDE 257            // coeffs row stride

// workspace layout (in floats)
#define WS_ROT   0                          // 64 * 9
#define WS_G     576                        // 64 * 27
#define WS_SHAPE 4096                       // BATCH * M3
#define WS_TEX   (4096 + (size_t)BATCH * M3)

// -------------------------------------------------------------------------
// Kernel 1: per-batch rotation matrices and SH gains
// rot = transpose(Rz @ Ry @ Rx); g[k][c] = gamma[c*9+k] + init_lit[k]
// -------------------------------------------------------------------------
__global__ void prep_kernel(const float* __restrict__ coeffs,
                            const float* __restrict__ init_lit,
                            float* __restrict__ ws) {
    int b = threadIdx.x;
    if (b >= BATCH) return;
    const float* c = coeffs + b * CSTRIDE;
    float ax = c[224], ay = c[225], az = c[226];
    float cx = cosf(ax), sx = sinf(ax);
    float cy = cosf(ay), sy = sinf(ay);
    float cz = cosf(az), sz = sinf(az);
    // M = Rz @ Ry @ Rx
    float m00 = cz * cy, m01 = cz * sy * sx - sz * cx, m02 = cz * sy * cx + sz * sx;
    float m10 = sz * cy, m11 = sz * sy * sx + cz * cx, m12 = sz * sy * cx - cz * sx;
    float m20 = -sy,     m21 = cy * sx,                m22 = cy * cx;
    float* rot = ws + WS_ROT + b * 9;   // rot = M^T, row-major
    rot[0] = m00; rot[1] = m10; rot[2] = m20;
    rot[3] = m01; rot[4] = m11; rot[5] = m21;
    rot[6] = m02; rot[7] = m12; rot[8] = m22;
    float* g = ws + WS_G + b * 27;
    for (int k = 0; k < 9; ++k) {
        float il = init_lit[k];
        for (int ch = 0; ch < 3; ++ch)
            g[k * 3 + ch] = c[227 + ch * 9 + k] + il;
    }
}

// -------------------------------------------------------------------------
// Kernel 2: fused WMMA GEMMs with LDS-transposed, coalesced epilogue
//   shape[b][m] = idBase[m]·id_c[b] + exBase[m]·ex_c[b] + meanshape[m]
//   tex  [b][m] = (texBase[m]·tex_c[b] + meantex[m]) / 255
// One wave per 16(M)x16(batch) tile; V_WMMA_F32_16X16X4_F32 over K.
// A 16x4 layout: lanes 0-15 carry (K=k,k+1), lanes 16-31 carry (K=k+2,k+3).
// C/D layout: VGPR r, lane L -> M = r + (L>=16 ? 8 : 0), Ncol = L & 15.
// Epilogue: tile -> LDS (17-float padded rows) -> contiguous-in-m float4
// stores, so each 64B sector is written exactly once.
// -------------------------------------------------------------------------
#define LDS_PITCH 17

__global__ __launch_bounds__(128)
void gemm_kernel(const float* __restrict__ coeffs,
                 const float* __restrict__ meanshape,
                 const float* __restrict__ idBase,
                 const float* __restrict__ exBase,
                 const float* __restrict__ meantex,
                 const float* __restrict__ texBase,
                 float* __restrict__ ws) {
    __shared__ float lsS[BATCH * LDS_PITCH];   // [b][m_local]
    __shared__ float lsT[BATCH * LDS_PITCH];

    const int lane  = threadIdx.x & 31;
    const int bcol  = threadIdx.x >> 5;          // 0..3 : 16-wide batch tile
    const int m0    = blockIdx.x * 16;

    const int kHalf = (lane >> 4) << 1;          // 0 or 2
    int mRow = m0 + (lane & 15);
    if (mRow > M3 - 1) mRow = M3 - 1;            // clamp (no divergence)
    const int bIdx = bcol * 16 + (lane & 15);    // batch column (< 64)

    const float* Aid = idBase  + (size_t)mRow * 80;
    const float* Aex = exBase  + (size_t)mRow * 64;
    const float* Atx = texBase + (size_t)mRow * 80;
    const float* Brw = coeffs + (size_t)bIdx * CSTRIDE;

    v8f accS = {};  // shape accumulator
    v8f accT = {};  // texture accumulator

    // ---- idBase @ id_c  (K = 80, coeff offset 0) ----
    __builtin_prefetch(Aex, 0, 1);
    for (int k = 0; k < 80; k += 4) {
        v2f a, b;
        a.x = Aid[k + kHalf];     a.y = Aid[k + kHalf + 1];
        b.x = Brw[k + kHalf];     b.y = Brw[k + kHalf + 1];
        accS = __builtin_amdgcn_wmma_f32_16x16x4_f32(
            false, a, false, b, (short)0, accS, false, false);
    }
    // ---- exBase @ ex_c  (K = 64, coeff offset 80) ----
    __builtin_prefetch(Atx, 0, 1);
    for (int k = 0; k < 64; k += 4) {
        v2f a, b;
        a.x = Aex[k + kHalf];       a.y = Aex[k + kHalf + 1];
        b.x = Brw[80 + k + kHalf];  b.y = Brw[80 + k + kHalf + 1];
        accS = __builtin_amdgcn_wmma_f32_16x16x4_f32(
            false, a, false, b, (short)0, accS, false, false);
    }
    // ---- texBase @ tex_c  (K = 80, coeff offset 144) ----
    for (int k = 0; k < 80; k += 4) {
        v2f a, b;
        a.x = Atx[k + kHalf];        a.y = Atx[k + kHalf + 1];
        b.x = Brw[144 + k + kHalf];  b.y = Brw[144 + k + kHalf + 1];
        accT = __builtin_amdgcn_wmma_f32_16x16x4_f32(
            false, a, false, b, (short)0, accT, false, false);
    }

    // ---- epilogue phase 1: accumulators -> LDS (transpose staging) ----
    const int rBase = (lane >> 4) * 8;           // M offset 0 or 8
    for (int r = 0; r < 8; ++r) {
        lsS[bIdx * LDS_PITCH + rBase + r] = accS[r];
        lsT[bIdx * LDS_PITCH + rBase + r] = accT[r];
    }
    __syncthreads();

    // ---- epilogue phase 2: coalesced contiguous-in-m stores ----
    // thread t owns (b = t>>1, m_local = (t&1)*8 .. +7): two float4 per array
    const int t  = threadIdx.x;
    const int bo = t >> 1;                       // batch row 0..63
    const int mh = (t & 1) * 8;                  // m_local half: 0 or 8
    float* shape_ws = ws + WS_SHAPE + (size_t)bo * M3;
    float* tex_ws   = ws + WS_TEX   + (size_t)bo * M3;
    const float* sRow = &lsS[bo * LDS_PITCH + mh];
    const float* tRow = &lsT[bo * LDS_PITCH + mh];

    if (m0 + 16 <= M3) {                          // full tile (all but last block)
        v4f s0, s1, t0, t1;
        for (int i = 0; i < 4; ++i) {
            int m = m0 + mh + i;
            s0[i] = sRow[i]     + meanshape[m];
            s1[i] = sRow[i + 4] + meanshape[m + 4];
            t0[i] = (tRow[i]     + meantex[m])     * (1.0f / 255.0f);
            t1[i] = (tRow[i + 4] + meantex[m + 4]) * (1.0f / 255.0f);
        }
        *(v4f*)(shape_ws + m0 + mh)     = s0;
        *(v4f*)(shape_ws + m0 + mh + 4) = s1;
        *(v4f*)(tex_ws   + m0 + mh)     = t0;
        *(v4f*)(tex_ws   + m0 + mh + 4) = t1;
    } else {                                      // ragged last tile
        for (int i = 0; i < 8; ++i) {
            int m = m0 + mh + i;
            if (m < M3) {
                shape_ws[m] = sRow[i] + meanshape[m];
                tex_ws[m]   = (tRow[i] + meantex[m]) * (1.0f / 255.0f);
            }
        }
    }
}

// -------------------------------------------------------------------------
// Kernel 3: per-vertex normals (recomputed from adjacent faces), SH color,
// rigid transform -> face_vertex and face_color.
// -------------------------------------------------------------------------
__global__ __launch_bounds__(256)
void vertex_kernel(const float* __restrict__ coeffs,
                   const int* __restrict__ point_buf,
                   const int* __restrict__ face_buf,
                   const float* __restrict__ ws,
                   float* __restrict__ out) {
    int gid = blockIdx.x * 256 + threadIdx.x;
    if (gid >= BATCH * NV) return;
    int b = gid / NV;
    int v = gid - b * NV;

    const float* shape = ws + WS_SHAPE + (size_t)b * M3;
    const float* texb  = ws + WS_TEX   + (size_t)b * M3;
    const float* rot   = ws + WS_ROT + b * 9;
    const float* g     = ws + WS_G + b * 27;
    const float* c     = coeffs + b * CSTRIDE;

    // ---- vertex normal: sum of unit face normals over up to 8 faces ----
    float vnx = 0.f, vny = 0.f, vnz = 0.f;
    for (int j = 0; j < 8; ++j) {
        int f = point_buf[v * 8 + j];
        if (f < NF) {
            int i0 = face_buf[f * 3 + 0];
            int i1 = face_buf[f * 3 + 1];
            int i2 = face_buf[f * 3 + 2];
            float p0x = shape[3 * i0], p0y = shape[3 * i0 + 1], p0z = shape[3 * i0 + 2];
            float p1x = shape[3 * i1], p1y = shape[3 * i1 + 1], p1z = shape[3 * i1 + 2];
            float p2x = shape[3 * i2], p2y = shape[3 * i2 + 1], p2z = shape[3 * i2 + 2];
            float e1x = p0x - p1x, e1y = p0y - p1y, e1z = p0z - p1z;
            float e2x = p1x - p2x, e2y = p1y - p2y, e2z = p1z - p2z;
            float crx = e1y * e2z - e1z * e2y;
            float cry = e1z * e2x - e1x * e2z;
            float crz = e1x * e2y - e1y * e2x;
            float n = fmaxf(sqrtf(crx * crx + cry * cry + crz * crz), 1e-12f);
            float inv = 1.0f / n;
            vnx += crx * inv; vny += cry * inv; vnz += crz * inv;
        }
    }
    float nn  = fmaxf(sqrtf(vnx * vnx + vny * vny + vnz * vnz), 1e-12f);
    float inv = 1.0f / nn;
    vnx *= inv; vny *= inv; vnz *= inv;

    // rotated normal: n_r = vn @ rot
    float nx = vnx * rot[0] + vny * rot[3] + vnz * rot[6];
    float ny = vnx * rot[1] + vny * rot[4] + vnz * rot[7];
    float nz = vnx * rot[2] + vny * rot[5] + vnz * rot[8];

    // rigid transform of this vertex: s @ rot + trans
    float sx = shape[3 * v], sy = shape[3 * v + 1], sz = shape[3 * v + 2];
    float tx = sx * rot[0] + sy * rot[3] + sz * rot[6] + c[254];
    float ty = sx * rot[1] + sy * rot[4] + sz * rot[7] + c[255];
    float tz = sx * rot[2] + sy * rot[5] + sz * rot[8] + c[256];

    size_t o = ((size_t)b * NV + v) * 3;
    out[o + 0] = tx;
    out[o + 1] = ty;
    out[o + 2] = 10.0f - tz;    // CAMERA_DISTANCE - z

    // ---- SH lighting ----
    // a0*c0 = 0.8862269; a1*c1 = 1.7724539; a2*c2 = 2.4270394
    float Y0 = 0.88622693f;
    float Y1 = -1.77245385f * ny;
    float Y2 =  1.77245385f * nz;
    float Y3 = -1.77245385f * nx;
    float Y4 =  2.42703939f * nx * ny;
    float Y5 = -2.42703939f * ny * nz;
    float Y6 =  0.70062926f * (3.0f * nz * nz - 1.0f);
    float Y7 = -2.42703939f * nx * nz;
    float Y8 =  1.21351970f * (nx * nx - ny * ny);

    size_t oc = (size_t)BATCH * NV * 3 + o;
    for (int ch = 0; ch < 3; ++ch) {
        float r = Y0 * g[0 * 3 + ch] + Y1 * g[1 * 3 + ch] + Y2 * g[2 * 3 + ch]
                + Y3 * g[3 * 3 + ch] + Y4 * g[4 * 3 + ch] + Y5 * g[5 * 3 + ch]
                + Y6 * g[6 * 3 + ch] + Y7 * g[7 * 3 + ch] + Y8 * g[8 * 3 + ch];
        out[oc + ch] = r * texb[3 * v + ch];
    }
}

// -------------------------------------------------------------------------
// Kernel 4: landmarks = perspective projection of face_vertex at keypoints
// -------------------------------------------------------------------------
__global__ void landmark_kernel(const float* __restrict__ persc,
                                const int* __restrict__ keypoints,
                                const float* __restrict__ face_vertex,
                                float* __restrict__ out_lm) {
    int t = blockIdx.x * blockDim.x + threadIdx.x;
    if (t >= BATCH * 68) return;
    int b = t / 68;
    int k = t - b * 68;
    int v = keypoints[k];
    const float* fv = face_vertex + ((size_t)b * NV + v) * 3;
    float x = fv[0], y = fv[1], z = fv[2];
    float p0 = x * persc[0] + y * persc[3] + z * persc[6];
    float p1 = x * persc[1] + y * persc[4] + z * persc[7];
    float p2 = x * persc[2] + y * persc[5] + z * persc[8];
    float invz = 1.0f / p2;
    out_lm[t * 2 + 0] = p0 * invz;
    out_lm[t * 2 + 1] = p1 * invz;
}

// -------------------------------------------------------------------------
extern "C" void kernel_launch(void* const* d_in, const int* in_sizes, int n_in,
                              void* d_out, int out_size, void* d_ws, size_t ws_size,
                              hipStream_t stream) {
    (void)in_sizes; (void)n_in; (void)out_size; (void)ws_size;

    const float* coeffs     = (const float*)d_in[0];
    const float* meanshape  = (const float*)d_in[1];
    const float* idBase     = (const float*)d_in[2];
    const float* exBase     = (const float*)d_in[3];
    const float* meantex    = (const float*)d_in[4];
    const float* texBase    = (const float*)d_in[5];
    const float* persc_proj = (const float*)d_in[6];
    const float* init_lit   = (const float*)d_in[7];
    const int*   point_buf  = (const int*)d_in[8];
    const int*   face_buf   = (const int*)d_in[9];
    const int*   keypoints  = (const int*)d_in[10];

    float* out = (float*)d_out;
    float* ws  = (float*)d_ws;

    // 1) rotations + SH gains
    prep_kernel<<<1, 64, 0, stream>>>(coeffs, init_lit, ws);

    // 2) fused WMMA GEMMs: shape + texture
    int mtiles = (M3 + 15) / 16;   // 6696
    gemm_kernel<<<mtiles, 128, 0, stream>>>(coeffs, meanshape, idBase, exBase,
                                            meantex, texBase, ws);

    // 3) normals + color + transform
    int nthreads = BATCH * NV;
    vertex_kernel<<<(nthreads + 255) / 256, 256, 0, stream>>>(
        coeffs, point_buf, face_buf, ws, out);

    // 4) landmarks (reads face_vertex portion of out)
    float* out_lm = out + (size_t)2 * BATCH * NV * 3;
    landmark_kernel<<<(BATCH * 68 + 255) / 256, 256, 0, stream>>>(
        persc_proj, keypoints, out, out_lm);
}